// Prior_SPDE_32384053412564
// MI455X (gfx1250) — compile-verified
//
#include <hip/hip_runtime.h>

typedef __attribute__((ext_vector_type(2))) float v2f;
typedef __attribute__((ext_vector_type(8))) float v8f;

namespace {
constexpr int NBN  = 1024;           // nodes per time slice (32x32)
constexpr int NT   = 9;              // time slices
constexpr int TNQ  = NBN * NT;       // 9216  (Q side)
constexpr int NTI  = TNQ / 16;       // 576 tiles per side
constexpr int NWAVE = NTI * NTI;     // 331776 tiles == waves
}

// ---------------------------------------------------------------------------
// Kernel 1: build the 8 dense invM_k = I + A_k blocks (1024x1024 each) in ws.
// 9-point stencil, Dirichlet BC, DX=DY=DT=1. Row-wise coefficients.
// ---------------------------------------------------------------------------
__global__ void build_invM_kernel(const float* __restrict__ kappa,
                                  const float* __restrict__ mvec,
                                  const float* __restrict__ Hmat,
                                  float* __restrict__ invM)
{
    int idx = blockIdx.x * blockDim.x + threadIdx.x;   // exactly 8*1024*1024
    int k   = idx >> 20;          // invM block 0..7 (time slice k+1)
    int rem = idx & 0xFFFFF;
    int r   = rem >> 10;          // row node
    int c   = rem & 1023;         // col node
    int t   = k + 1;
    int iy  = r >> 5, ix = r & 31;

    float kap = kappa[r * NT + t];
    float k2  = kap * kap;
    float m1  = mvec[r * NT + t];
    float m2  = mvec[(NBN + r) * NT + t];
    float H11 = Hmat[r * NT + t];                 // H[0,0]
    float H12 = Hmat[(NBN + r) * NT + t];         // H[0,1]
    float H22 = Hmat[(3 * NBN + r) * NT + t];     // H[1,1]

    float val = 0.0f;
    int d = c - r;
    if (d == 0)                                   val = 1.0f + k2 + 2.0f*H11 + 2.0f*H22;
    else if (d ==  1  && ix < 31)                 val =  0.5f*m1 - H11;
    else if (d == -1  && ix > 0)                  val = -0.5f*m1 - H11;
    else if (d ==  32 && iy < 31)                 val =  0.5f*m2 - H22;
    else if (d == -32 && iy > 0)                  val = -0.5f*m2 - H22;
    else if (d ==  33 && ix < 31 && iy < 31)      val = -0.5f*H12;   // (+1,+1)
    else if (d == -31 && ix < 31 && iy > 0)       val =  0.5f*H12;   // (+1,-1)
    else if (d ==  31 && ix > 0  && iy < 31)      val =  0.5f*H12;   // (-1,+1)
    else if (d == -33 && ix > 0  && iy > 0)       val = -0.5f*H12;   // (-1,-1)

    invM[(k << 20) + (r << 10) + c] = val;
}

// ---------------------------------------------------------------------------
// WMMA fragment loaders for V_WMMA_F32_16X16X4_F32 (ISA 7.12.2 layouts).
// A (16x4): lanes 0-15 -> M=lane, VGPR0/1 = K {2*hi, 2*hi+1}. B mirrored.
// ld = 1024 for all invM blocks. rowb/colb/kb are wave-uniform (SGPR).
// ---------------------------------------------------------------------------
__device__ __forceinline__ v2f ldA(const float* M, int rowb, int kb, int l16, int hi) {
    const float* p = M + (rowb + l16) * NBN + kb + 2 * hi;     // contiguous pair
    v2f a; a.x = p[0]; a.y = p[1]; return a;
}
__device__ __forceinline__ v2f ldB(const float* M, int colb, int kb, int l16, int hi) {
    const float* p = M + (kb + 2 * hi) * NBN + colb + l16;     // stride-ld pair
    v2f b; b.x = p[0]; b.y = p[NBN]; return b;
}
__device__ __forceinline__ v2f ldAT(const float* M, int rowb, int kb, int l16, int hi) {
    const float* p = M + (kb + 2 * hi) * NBN + rowb + l16;     // M^T as A
    v2f a; a.x = p[0]; a.y = p[NBN]; return a;
}
__device__ __forceinline__ v2f ldBT(const float* M, int colb, int kb, int l16, int hi) {
    const float* p = M + (colb + l16) * NBN + kb + 2 * hi;     // M^T as B
    v2f b; b.x = p[0]; b.y = p[1]; return b;
}

// ---------------------------------------------------------------------------
// Kernel 2: one wave32 per 16x16 tile of Q (9216x9216). Tile coordinates are
// forced into SGPRs via readfirstlane so all branches / loop bounds are
// scalar: EXEC provably stays all-ones across the WMMA chains, and address
// bases live in SALU.
// ---------------------------------------------------------------------------
__global__ __launch_bounds__(256) void assemble_Q_kernel(
        const float* __restrict__ invM, float* __restrict__ out)
{
    int wave0 = blockIdx.x * (blockDim.x >> 5) + (threadIdx.x >> 5);
    // wave-uniform by construction; make it compiler-visible (SGPR)
    int wave = __builtin_amdgcn_readfirstlane(wave0);
    if (wave >= NWAVE) return;
    int lane = threadIdx.x & 31;
    int hi   = lane >> 4;        // lane half
    int l16  = lane & 15;

    int TR = wave / NTI;         // tile row  0..575   (SGPR)
    int TC = wave % NTI;         // tile col  0..575   (SGPR)
    int bi = TR >> 6, bj = TC >> 6;   // 9x9 time-block index
    int I  = TR & 63, J = TC & 63;    // tile within 64x64 block

    v8f acc = {0.f,0.f,0.f,0.f,0.f,0.f,0.f,0.f};

    if (bi == bj) {
        int kb = (bi == 0) ? 0 : (bi - 1);
        const float* M = invM + ((long)kb << 20);
        int dIJ = (I > J) ? (I - J) : (J - I);
        if (dIJ <= 6) {                          // banded: product tiles vanish beyond
            int Klo = ((I > J) ? I : J) - 3; if (Klo < 0)  Klo = 0;
            int Khi = ((I < J) ? I : J) + 3; if (Khi > 63) Khi = 63;
            if (bi == 0) {
                // C = invM0^T * invM0  (banded; 4 chained wmma per 16-wide K tile)
                v8f c = {0.f,0.f,0.f,0.f,0.f,0.f,0.f,0.f};
                for (int K = Klo; K <= Khi; ++K) {
                    int kb16 = K * 16;
                    #pragma unroll
                    for (int k0 = 0; k0 < 16; k0 += 4) {
                        v2f a = ldAT(M, I * 16, kb16 + k0, l16, hi);
                        v2f b = ldB (M, J * 16, kb16 + k0, l16, hi);
                        c = __builtin_amdgcn_wmma_f32_16x16x4_f32(
                                false, a, false, b, (short)0, c, false, false);
                    }
                }
                // Q00 = A0^T A0 + 1.05 I = invM0^T invM0 - invM0 - invM0^T + 2.05 I
                #pragma unroll
                for (int v = 0; v < 8; ++v) {
                    int Mr = hi ? (v + 8) : v;
                    int rL = I * 16 + Mr, cL = J * 16 + l16;
                    float val = c[v] - M[rL * NBN + cL] - M[cL * NBN + rL];
                    if (rL == cL) val += 2.05f;
                    acc[v] = val;
                }
            } else {
                // c1 = tile of invM*invM ; c2 = tile of invM^T*invM^T == (MM)^T tile
                v8f c1 = {0.f,0.f,0.f,0.f,0.f,0.f,0.f,0.f};
                v8f c2 = {0.f,0.f,0.f,0.f,0.f,0.f,0.f,0.f};
                for (int K = Klo; K <= Khi; ++K) {
                    int kb16 = K * 16;
                    #pragma unroll
                    for (int k0 = 0; k0 < 16; k0 += 4) {
                        v2f a1 = ldA (M, I * 16, kb16 + k0, l16, hi);
                        v2f b1 = ldB (M, J * 16, kb16 + k0, l16, hi);
                        c1 = __builtin_amdgcn_wmma_f32_16x16x4_f32(
                                false, a1, false, b1, (short)0, c1, false, false);
                        v2f a2 = ldAT(M, I * 16, kb16 + k0, l16, hi);
                        v2f b2 = ldBT(M, J * 16, kb16 + k0, l16, hi);
                        c2 = __builtin_amdgcn_wmma_f32_16x16x4_f32(
                                false, a2, false, b2, (short)0, c2, false, false);
                    }
                }
                float dadd = (bi < 8) ? 1.05f : 0.05f;   // +I only for blocks 1..7
                #pragma unroll
                for (int v = 0; v < 8; ++v) {
                    int Mr = hi ? (v + 8) : v;
                    int rL = I * 16 + Mr, cL = J * 16 + l16;
                    float val = 0.5f * (c1[v] + c2[v]);
                    if (rL == cL) val += dadd;
                    acc[v] = val;
                }
            }
        }
    } else if (bi - bj == 1 || bj - bi == 1) {
        int idx = (bi < bj) ? bi : bj;
        const float* M = invM + ((long)idx << 20);
        int dIJ = (I > J) ? (I - J) : (J - I);
        if (dIJ <= 3) {                          // invM band: |I-J| <= 3 tiles
            #pragma unroll
            for (int v = 0; v < 8; ++v) {
                int Mr = hi ? (v + 8) : v;
                int rL = I * 16 + Mr, cL = J * 16 + l16;
                acc[v] = -0.5f * (M[rL * NBN + cL] + M[cL * NBN + rL]);
            }
        }
    }
    // every tile writes (zeros outside the block-tridiagonal band)
    float* obase = out + (long)(TR * 16) * TNQ + TC * 16;
    #pragma unroll
    for (int v = 0; v < 8; ++v) {
        int Mr = hi ? (v + 8) : v;
        obase[(long)Mr * TNQ + l16] = acc[v];
    }
}

extern "C" void kernel_launch(void* const* d_in, const int* in_sizes, int n_in,
                              void* d_out, int out_size, void* d_ws, size_t ws_size,
                              hipStream_t stream) {
    const float* kappa = (const float*)d_in[0];   // [1,1,1024,9]
    const float* mvec  = (const float*)d_in[1];   // [1,2,1024,9]
    const float* Hmat  = (const float*)d_in[2];   // [1,2,2,1024,9]
    // d_in[3] = tau : unused by the reference output
    float* out  = (float*)d_out;                  // [1,9216,9216] f32
    float* invM = (float*)d_ws;                   // needs 8 * 1024*1024 * 4B = 32 MiB
    (void)in_sizes; (void)n_in; (void)out_size; (void)ws_size;

    // 1) dense invM blocks into scratch
    build_invM_kernel<<<(8 * 1024 * 1024) / 256, 256, 0, stream>>>(kappa, mvec, Hmat, invM);

    // 2) one wave per 16x16 tile of Q; 8 waves per 256-thread block
    assemble_Q_kernel<<<NWAVE / 8, 256, 0, stream>>>(invM, out);
}